// LPModelGNNProductEuclidean_70626442215706
// MI455X (gfx1250) — compile-verified
//
#include <hip/hip_runtime.h>
#include <hip/hip_bf16.h>

typedef __attribute__((ext_vector_type(2))) float v2f;
typedef __attribute__((ext_vector_type(8))) float v8f;

// ---------------------------------------------------------------------------
// Degree / normalization helpers
// ---------------------------------------------------------------------------
__global__ void fill_ones_kernel(float* __restrict__ p, int n) {
  int i = blockIdx.x * blockDim.x + threadIdx.x;
  if (i < n) p[i] = 1.0f;  // self-loop contributes 1 to every degree
}

__global__ void count_deg_kernel(const int* __restrict__ dst,
                                 float* __restrict__ deg, int E) {
  int e = blockIdx.x * blockDim.x + threadIdx.x;
  if (e < E) unsafeAtomicAdd(&deg[dst[e]], 1.0f);  // global_atomic_add_f32
}

__global__ void finalize_dinv_kernel(float* __restrict__ p, int n) {
  int i = blockIdx.x * blockDim.x + threadIdx.x;
  if (i < n) p[i] = rsqrtf(p[i]);  // deg >= 1 always (self loop)
}

// ---------------------------------------------------------------------------
// WMMA GEMM: Y[M,64] = act(X[M,K]) @ W[K,64], fp32, V_WMMA_F32_16X16X4_F32.
// One wave computes one 16x16 tile; 4 waves/block span N=64.
// Requires M % 16 == 0 (100000 = 6250*16).
// ---------------------------------------------------------------------------
template <int K, bool RELU>
__global__ void gemm_n64_kernel(const float* __restrict__ X,
                                const float* __restrict__ W,
                                float* __restrict__ Y) {
  const int mtile = blockIdx.x;            // 16-row tile index
  const int wave  = threadIdx.x >> 5;      // 0..3 -> 16-col tile
  const int lane  = threadIdx.x & 31;
  const int half  = lane >> 4;             // 0: lanes 0-15, 1: lanes 16-31
  const int l15   = lane & 15;
  const int khalf = half << 1;             // 0 or 2 (K sub-offset)
  const int col   = (wave << 4) + l15;     // output column 0..63

  const float* Arow = X + (size_t)(mtile * 16 + l15) * K;
  const float* Wc   = W + col;

  v8f c = {};
  for (int k = 0; k < K; k += 4) {
    float a0 = Arow[k + khalf];
    float a1 = Arow[k + khalf + 1];
    if (RELU) { a0 = fmaxf(a0, 0.0f); a1 = fmaxf(a1, 0.0f); }
    v2f a, b;
    a.x = a0;
    a.y = a1;
    b.x = Wc[(size_t)(k + khalf) * 64];
    b.y = Wc[(size_t)(k + khalf + 1) * 64];
    // (neg_a, A, neg_b, B, c_mod, C, reuse_a, reuse_b)
    c = __builtin_amdgcn_wmma_f32_16x16x4_f32(false, a, false, b,
                                              (short)0, c, false, false);
  }

  // C/D layout: VGPR r -> M=r (lanes 0-15), M=r+8 (lanes 16-31); N = lane&15.
  const int mbase = mtile * 16 + (half << 3);
#pragma unroll
  for (int r = 0; r < 8; ++r) {
    Y[(size_t)(mbase + r) * 64 + col] = c[r];
  }
}

// ---------------------------------------------------------------------------
// out[v,f] = xw[v,f] * dinv[v]^2 + bias[f]   (self-loop term + bias)
// ---------------------------------------------------------------------------
__global__ void self_bias_kernel(const float* __restrict__ xw,
                                 const float* __restrict__ bias,
                                 const float* __restrict__ dinv,
                                 float* __restrict__ out, int total) {
  int gid = blockIdx.x * blockDim.x + threadIdx.x;
  if (gid >= total) return;
  int v = gid >> 6;
  int f = gid & 63;
  float di = dinv[v];
  out[gid] = xw[gid] * di * di + bias[f];
}

// ---------------------------------------------------------------------------
// Edge scatter: out[dst] += xw[src] * dinv[src]*dinv[dst].
// 16 threads per edge, float4 gather each (256B contiguous per edge),
// 4 hardware fp32 atomics per thread. L2-resident on MI455X (192MB).
// ---------------------------------------------------------------------------
__global__ void edge_scatter_kernel(const float* __restrict__ xw,
                                    float* __restrict__ out,
                                    const int* __restrict__ src,
                                    const int* __restrict__ dst,
                                    const float* __restrict__ dinv, int E) {
  long long gid = (long long)blockIdx.x * blockDim.x + threadIdx.x;
  int e = (int)(gid >> 4);
  if (e >= E) return;
  int c = ((int)gid & 15) << 2;  // float offset within row: 0,4,...,60
  int s = src[e];
  int d = dst[e];
  float norm = dinv[s] * dinv[d];
  const float4 v = *(const float4*)(xw + (size_t)s * 64 + c);
  float* o = out + (size_t)d * 64 + c;
  unsafeAtomicAdd(o + 0, v.x * norm);
  unsafeAtomicAdd(o + 1, v.y * norm);
  unsafeAtomicAdd(o + 2, v.z * norm);
  unsafeAtomicAdd(o + 3, v.w * norm);
}

// ---------------------------------------------------------------------------
// Fermi-Dirac decoder: probs = 1 / (exp(||z1_i-z1_j||^2 + ||z2_i-z2_j||^2 - R) + 1)
// ---------------------------------------------------------------------------
__global__ void decode_kernel(const float* __restrict__ z1,
                              const float* __restrict__ z2,
                              const int* __restrict__ ei,
                              const int* __restrict__ ej,
                              float* __restrict__ out, int P) {
  int p = blockIdx.x * blockDim.x + threadIdx.x;
  if (p >= P) return;
  int i = ei[p];
  int j = ej[p];
  const float4* a1 = (const float4*)(z1 + (size_t)i * 64);
  const float4* b1 = (const float4*)(z1 + (size_t)j * 64);
  const float4* a2 = (const float4*)(z2 + (size_t)i * 64);
  const float4* b2 = (const float4*)(z2 + (size_t)j * 64);
  float s = 0.0f;
#pragma unroll
  for (int t = 0; t < 16; ++t) {
    float4 x = a1[t], y = b1[t];
    float dx = x.x - y.x, dy = x.y - y.y, dz = x.z - y.z, dw = x.w - y.w;
    s += dx * dx + dy * dy + dz * dz + dw * dw;
  }
#pragma unroll
  for (int t = 0; t < 16; ++t) {
    float4 x = a2[t], y = b2[t];
    float dx = x.x - y.x, dy = x.y - y.y, dz = x.z - y.z, dw = x.w - y.w;
    s += dx * dx + dy * dy + dz * dz + dw * dw;
  }
  // R = 2.0, T = 1.0
  out[p] = 1.0f / (__expf(s - 2.0f) + 1.0f);
}

// ---------------------------------------------------------------------------
// Host-side orchestration (all launches on `stream`; no allocs/syncs).
// ---------------------------------------------------------------------------
extern "C" void kernel_launch(void* const* d_in, const int* in_sizes, int n_in,
                              void* d_out, int out_size, void* d_ws, size_t ws_size,
                              hipStream_t stream) {
  const float* X    = (const float*)d_in[0];   // [N,128]
  const int*   edge = (const int*)d_in[1];     // [2,E] row-major
  const int*   eli  = (const int*)d_in[2];     // [2,P] row-major
  const float* W1_1 = (const float*)d_in[3];
  const float* b1_1 = (const float*)d_in[4];
  const float* W2_1 = (const float*)d_in[5];
  const float* b2_1 = (const float*)d_in[6];
  const float* W1_2 = (const float*)d_in[7];
  const float* b1_2 = (const float*)d_in[8];
  const float* W2_2 = (const float*)d_in[9];
  const float* b2_2 = (const float*)d_in[10];

  const int N = in_sizes[0] / 128;   // 100000 nodes
  const int E = in_sizes[1] / 2;     // 1600000 edges
  const int P = in_sizes[2] / 2;     // 500000 label pairs
  const int* src = edge;
  const int* dst = edge + E;
  const int* ei  = eli;
  const int* ej  = eli + P;

  // Workspace layout (floats): dinv (padded) + 4 node-feature buffers.
  float*       ws   = (float*)d_ws;
  float*       dinv = ws;
  const size_t PAD  = 102400;            // N rounded up, keeps buffers 16B-aligned
  const size_t NB   = (size_t)N * 64;    // 6.4M floats per buffer
  float* buf0 = ws + PAD;                // xw branch 1
  float* buf1 = buf0 + NB;               // xw branch 2
  float* buf2 = buf1 + NB;               // h1 / z1
  float* buf3 = buf2 + NB;               // h2 / z2

  const int TB = 256;
  const int total  = N * 64;
  const dim3 gemm_grid(N / 16);          // 6250 tiles (N % 16 == 0)
  const dim3 gemm_block(128);            // 4 waves -> 4 N-tiles
  const int node_blocks  = (N + TB - 1) / TB;
  const int feat_blocks  = (total + TB - 1) / TB;
  const int edge_blocks  = (E + TB - 1) / TB;
  const long long sc_thr = (long long)E * 16;
  const int scat_blocks  = (int)((sc_thr + TB - 1) / TB);
  const int pair_blocks  = (P + TB - 1) / TB;

  // --- degree normalization: dinv = rsqrt(1 + in-degree) -------------------
  fill_ones_kernel<<<node_blocks, TB, 0, stream>>>(dinv, N);
  count_deg_kernel<<<edge_blocks, TB, 0, stream>>>(dst, dinv, E);
  finalize_dinv_kernel<<<node_blocks, TB, 0, stream>>>(dinv, N);

  // --- layer 1: xw = X @ W (WMMA), then out = D^-1/2 A D^-1/2 xw + b -------
  gemm_n64_kernel<128, false><<<gemm_grid, gemm_block, 0, stream>>>(X, W1_1, buf0);
  gemm_n64_kernel<128, false><<<gemm_grid, gemm_block, 0, stream>>>(X, W2_1, buf1);
  self_bias_kernel<<<feat_blocks, TB, 0, stream>>>(buf0, b1_1, dinv, buf2, total);
  self_bias_kernel<<<feat_blocks, TB, 0, stream>>>(buf1, b2_1, dinv, buf3, total);
  edge_scatter_kernel<<<scat_blocks, TB, 0, stream>>>(buf0, buf2, src, dst, dinv, E);
  edge_scatter_kernel<<<scat_blocks, TB, 0, stream>>>(buf1, buf3, src, dst, dinv, E);
  // buf2/buf3 now hold pre-ReLU h1/h2; ReLU fused into next GEMM's A load.

  // --- layer 2 -------------------------------------------------------------
  gemm_n64_kernel<64, true><<<gemm_grid, gemm_block, 0, stream>>>(buf2, W1_2, buf0);
  gemm_n64_kernel<64, true><<<gemm_grid, gemm_block, 0, stream>>>(buf3, W2_2, buf1);
  self_bias_kernel<<<feat_blocks, TB, 0, stream>>>(buf0, b1_2, dinv, buf2, total);
  self_bias_kernel<<<feat_blocks, TB, 0, stream>>>(buf1, b2_2, dinv, buf3, total);
  edge_scatter_kernel<<<scat_blocks, TB, 0, stream>>>(buf0, buf2, src, dst, dinv, E);
  edge_scatter_kernel<<<scat_blocks, TB, 0, stream>>>(buf1, buf3, src, dst, dinv, E);
  // buf2 = z1, buf3 = z2

  // --- Fermi-Dirac decode --------------------------------------------------
  decode_kernel<<<pair_blocks, TB, 0, stream>>>(buf2, buf3, ei, ej, (float*)d_out, P);
}